// TemplateLayer_4337916969171
// MI455X (gfx1250) — compile-verified
//
#include <hip/hip_runtime.h>
#include <hip/hip_bf16.h>

typedef __attribute__((ext_vector_type(2))) float v2f;
typedef __attribute__((ext_vector_type(8))) float v8f;

#define N_EDGES_C 750000

// ---------------------------------------------------------------------------
// XW = X (nrows x 64) @ W (64 x 64), fp32 WMMA 16x16x4.
// 128 threads = 4 waves/block; wave w computes rows [blk*64 + w*16, +16).
// A layout (16x4 f32): lane<16 -> M=lane, K={kb,kb+1}; lane>=16 -> M=lane-16,
// K={kb+2,kb+3}. B layout (4x16) mirrors A. C/D: VGPR i -> M=i (lanes 0-15),
// M=i+8 (lanes 16-31), N = lane&15.
//
// W is staged into LDS in B-fragment order so each lane's fragment
// (W[k,n], W[k+1,n]) is one contiguous 8-byte ds_load:
//   WlB[(k>>1)*128 + n*2 + (k&1)] = W[k*64 + n]
// ---------------------------------------------------------------------------
__global__ __launch_bounds__(128) void gemm64_wmma(
    const float* __restrict__ X, const float* __restrict__ W,
    float* __restrict__ XW, int nrows)
{
    __shared__ float WlB[64 * 64];
    for (int i = threadIdx.x; i < 64 * 64; i += 128) {
        const int k = i >> 6, n = i & 63;
        WlB[(k >> 1) * 128 + n * 2 + (k & 1)] = W[i];
    }
    __syncthreads();

    const int lane  = threadIdx.x & 31;
    const int wave  = threadIdx.x >> 5;
    const int row0  = (blockIdx.x * 4 + wave) * 16;
    if (row0 >= nrows) return;                 // wave-uniform: EXEC stays full

    const int mrow  = row0 + (lane & 15);
    const int khalf = (lane >> 4) * 2;         // 0 or 2
    const int ncol  = lane & 15;
    const float* xrow = X + (long long)mrow * 64;

    v8f acc0 = {}, acc1 = {}, acc2 = {}, acc3 = {};

    #pragma unroll
    for (int kb = 0; kb < 64; kb += 4) {
        const int k = kb + khalf;              // even
        const float* bk = WlB + (k >> 1) * 128;
        v2f a = *(const v2f*)(xrow + k);
        v2f b0 = *(const v2f*)(bk + ( 0 + ncol) * 2);
        v2f b1 = *(const v2f*)(bk + (16 + ncol) * 2);
        v2f b2 = *(const v2f*)(bk + (32 + ncol) * 2);
        v2f b3 = *(const v2f*)(bk + (48 + ncol) * 2);
        acc0 = __builtin_amdgcn_wmma_f32_16x16x4_f32(false, a, false, b0, (short)0, acc0, false, false);
        acc1 = __builtin_amdgcn_wmma_f32_16x16x4_f32(false, a, false, b1, (short)0, acc1, false, false);
        acc2 = __builtin_amdgcn_wmma_f32_16x16x4_f32(false, a, false, b2, (short)0, acc2, false, false);
        acc3 = __builtin_amdgcn_wmma_f32_16x16x4_f32(false, a, false, b3, (short)0, acc3, false, false);
    }

    const int rbase = row0 + (lane >> 4) * 8;
    #pragma unroll
    for (int i = 0; i < 8; ++i) {
        const long long r = (long long)(rbase + i) * 64;
        XW[r +  0 + ncol] = acc0[i];
        XW[r + 16 + ncol] = acc1[i];
        XW[r + 32 + ncol] = acc2[i];
        XW[r + 48 + ncol] = acc3[i];
    }
}

// ---------------------------------------------------------------------------
// deg[dst[i]] += vals[i]
// ---------------------------------------------------------------------------
__global__ void deg_kernel(const int* __restrict__ dst,
                           const float* __restrict__ vals,
                           float* deg, int nnz)
{
    int i = blockIdx.x * blockDim.x + threadIdx.x;
    if (i < nnz) atomicAdd(&deg[dst[i]], vals[i]);
}

// ---------------------------------------------------------------------------
// agg[dst[i], c] += vals[i] * xw[src[i], c]; 4 channels per thread.
// n = nnz * 16 threads total.
// ---------------------------------------------------------------------------
__global__ void scatter4(const float* __restrict__ xw,
                         const int* __restrict__ src,
                         const int* __restrict__ dst,
                         const float* __restrict__ vals,
                         float* agg, long long n)
{
    long long t = (long long)blockIdx.x * blockDim.x + threadIdx.x;
    if (t >= n) return;
    const long long i  = t >> 4;
    const int       c4 = (int)(t & 15) * 4;
    const float v = vals[i];
    const int   s = src[i];
    const int   d = dst[i];
    const float4 x = *(const float4*)(xw + (long long)s * 64 + c4);
    float* a = agg + (long long)d * 64 + c4;
    atomicAdd(a + 0, v * x.x);
    atomicAdd(a + 1, v * x.y);
    atomicAdd(a + 2, v * x.z);
    atomicAdd(a + 3, v * x.w);
}

// ---------------------------------------------------------------------------
// out[t] = sigmoid(agg[t] / deg[t/64])
// ---------------------------------------------------------------------------
__global__ void norm_sigmoid(const float* __restrict__ agg,
                             const float* __restrict__ deg,
                             float* __restrict__ out, long long n)
{
    long long t = (long long)blockIdx.x * blockDim.x + threadIdx.x;
    if (t >= n) return;
    const float v = agg[t] / deg[t >> 6];
    out[t] = 1.0f / (1.0f + __expf(-v));
}

// ---------------------------------------------------------------------------
extern "C" void kernel_launch(void* const* d_in, const int* in_sizes, int n_in,
                              void* d_out, int out_size, void* d_ws, size_t ws_size,
                              hipStream_t stream)
{
    const float* x2   = (const float*)d_in[0];   // [n_faces, 64]
    const int*   rows = (const int*)  d_in[1];   // inc_rows (edges)  [nnz]
    const int*   cols = (const int*)  d_in[2];   // inc_cols (faces)  [nnz]
    const float* vals = (const float*)d_in[3];   // [nnz]
    const float* W1   = (const float*)d_in[4];   // [64,64]
    const float* W2   = (const float*)d_in[5];   // [64,64]
    float* out = (float*)d_out;

    const int n_faces = in_sizes[0] / 64;        // 500000 (multiple of 16)
    const int nnz     = in_sizes[1];             // 1500000
    const int n_edges = N_EDGES_C;               // 750000 (multiple of 16)

    // Carve workspace (256B-aligned slabs).
    char*  ws  = (char*)d_ws;
    size_t off = 0;
    auto carve = [&](size_t bytes) -> char* {
        char* p = ws + off;
        off += (bytes + 255) & ~(size_t)255;
        return p;
    };
    float* xw   = (float*)carve((size_t)n_edges * 64 * sizeof(float)); // both GEMM outputs
    float* agg1 = (float*)carve((size_t)n_edges * 64 * sizeof(float)); // also x_1 in place
    float* agg2 = (float*)carve((size_t)n_faces * 64 * sizeof(float));
    float* deg1 = (float*)carve((size_t)n_edges * sizeof(float));
    float* deg2 = (float*)carve((size_t)n_faces * sizeof(float));

    hipMemsetAsync(agg1, 0, (size_t)n_edges * 64 * sizeof(float), stream);
    hipMemsetAsync(agg2, 0, (size_t)n_faces * 64 * sizeof(float), stream);
    hipMemsetAsync(deg1, 0, (size_t)n_edges * sizeof(float), stream);
    hipMemsetAsync(deg2, 0, (size_t)n_faces * sizeof(float), stream);

    const long long nScat = (long long)nnz * 16;

    // ---- level 1: faces -> edges -------------------------------------------
    gemm64_wmma<<<(n_faces / 16 + 3) / 4, 128, 0, stream>>>(x2, W1, xw, n_faces);
    deg_kernel<<<(nnz + 255) / 256, 256, 0, stream>>>(rows, vals, deg1, nnz);
    scatter4<<<(unsigned)((nScat + 255) / 256), 256, 0, stream>>>(xw, cols, rows, vals, agg1, nScat);
    norm_sigmoid<<<(unsigned)(((long long)n_edges * 64 + 255) / 256), 256, 0, stream>>>(
        agg1, deg1, agg1, (long long)n_edges * 64);

    // ---- level 2: edges -> faces -------------------------------------------
    gemm64_wmma<<<(n_edges / 16 + 3) / 4, 128, 0, stream>>>(agg1, W2, xw, n_edges);
    deg_kernel<<<(nnz + 255) / 256, 256, 0, stream>>>(cols, vals, deg2, nnz);
    scatter4<<<(unsigned)((nScat + 255) / 256), 256, 0, stream>>>(xw, rows, cols, vals, agg2, nScat);
    norm_sigmoid<<<(unsigned)(((long long)n_faces * 64 + 255) / 256), 256, 0, stream>>>(
        agg2, deg2, out, (long long)n_faces * 64);
}